// GCN_7997229105681
// MI455X (gfx1250) — compile-verified
//
#include <hip/hip_runtime.h>

typedef float v2f __attribute__((ext_vector_type(2)));
typedef float v8f __attribute__((ext_vector_type(8)));

#define C_IN  128
#define C_HID 64
#define C_OUT 16

// ---------------------------------------------------------------------------
// degree / normalization helpers
// ---------------------------------------------------------------------------
__global__ void k_set_ones(float* __restrict__ deg, int n) {
    int i = blockIdx.x * blockDim.x + threadIdx.x;
    if (i < n) deg[i] = 1.0f;                 // self-loop contributes 1
}

__global__ void k_deg_accum(const int* __restrict__ dst, float* __restrict__ deg, int e) {
    int i = blockIdx.x * blockDim.x + threadIdx.x;
    if (i < e) atomicAdd(&deg[dst[i]], 1.0f);
}

__global__ void k_deg_to_dinv(float* __restrict__ deg, int n) {
    int i = blockIdx.x * blockDim.x + threadIdx.x;
    if (i < n) {
        float d = deg[i];
        deg[i] = (d > 0.0f) ? rsqrtf(d) : 0.0f;
    }
}

// ---------------------------------------------------------------------------
// GEMM1: h1[N,64] = x[N,128] @ W1[128,64], exact f32 via V_WMMA_F32_16X16X4_F32
// block = 128 threads = 4 waves; block owns a 16-node row tile staged in LDS;
// wave w owns 16-wide column tile w.
// ---------------------------------------------------------------------------
__global__ __launch_bounds__(128) void k_gemm1_wmma(
    const float* __restrict__ x, const float* __restrict__ W1,
    float* __restrict__ h1)
{
    __shared__ float xs[16 * 132];            // padded stride -> conflict-free b64 reads
    const int m0 = blockIdx.x * 16;
    const int t  = threadIdx.x;

    // cooperative, coalesced float4 load of the 16x128 A tile
    #pragma unroll
    for (int i = 0; i < 4; ++i) {
        int q   = t + i * 128;                // 0..511 float4 slots
        int row = q >> 5;                     // 32 float4 per row
        int c4  = (q & 31) << 2;
        const float4 v = *(const float4*)(x + (size_t)(m0 + row) * C_IN + c4);
        float* p = &xs[row * 132 + c4];
        p[0] = v.x; p[1] = v.y; p[2] = v.z; p[3] = v.w;
    }
    __syncthreads();

    const int lane = t & 31;
    const int wave = t >> 5;
    const int mrow = lane & 15;
    const int kb   = (lane >> 4) << 1;        // 0 or 2 (A/B K sub-pair per half-wave)
    const int ncol = wave * 16 + (lane & 15);

    v8f acc = {};
    #pragma unroll
    for (int kk = 0; kk < C_IN; kk += 4) {
        v2f a = *(const v2f*)&xs[mrow * 132 + kk + kb];   // ds_load_b64
        v2f b;
        b.x = W1[(kk + kb) * C_HID + ncol];               // W1 is L2-hot (32 KB)
        b.y = W1[(kk + kb + 1) * C_HID + ncol];
        acc = __builtin_amdgcn_wmma_f32_16x16x4_f32(
                  false, a, false, b, (short)0, acc, false, false);
    }

    const int mbase = (lane >> 4) << 3;       // C/D: rows g (lanes 0-15) / g+8 (16-31)
    #pragma unroll
    for (int g = 0; g < 8; ++g)
        h1[(size_t)(m0 + mbase + g) * C_HID + ncol] = acc[g];
}

// ---------------------------------------------------------------------------
// GEMM2: h2[N,16] = relu(agg1)[N,64] @ W2[64,16]; ReLU fused into A-frag load.
// 8 waves per block, one 16-node tile per wave (uniform per-wave guard keeps
// EXEC all-ones for WMMA).
// ---------------------------------------------------------------------------
__global__ __launch_bounds__(256) void k_gemm2_wmma(
    const float* __restrict__ agg1, const float* __restrict__ W2,
    float* __restrict__ h2, int n_tiles)
{
    const int wave = threadIdx.x >> 5;
    const int tile = blockIdx.x * 8 + wave;
    if (tile >= n_tiles) return;              // whole-wave exit, EXEC stays uniform

    const int lane = threadIdx.x & 31;
    const int mrow = lane & 15;
    const int kb   = (lane >> 4) << 1;
    const int ncol = lane & 15;
    const int m0   = tile * 16;

    v8f acc = {};
    #pragma unroll
    for (int kk = 0; kk < C_HID; kk += 4) {
        v2f a = *(const v2f*)(agg1 + (size_t)(m0 + mrow) * C_HID + kk + kb);
        a.x = fmaxf(a.x, 0.0f);               // fused ReLU
        a.y = fmaxf(a.y, 0.0f);
        v2f b;
        b.x = W2[(kk + kb) * C_OUT + ncol];
        b.y = W2[(kk + kb + 1) * C_OUT + ncol];
        acc = __builtin_amdgcn_wmma_f32_16x16x4_f32(
                  false, a, false, b, (short)0, acc, false, false);
    }

    const int mbase = (lane >> 4) << 3;
    #pragma unroll
    for (int g = 0; g < 8; ++g)
        h2[(size_t)(m0 + mbase + g) * C_OUT + ncol] = acc[g];
}

// ---------------------------------------------------------------------------
// agg[i][c] = h[i][c] * dinv[i]^2 + bias[c]   (self-loop message + bias)
// also serves as the d_out initializer (overwrites harness poison).
// ---------------------------------------------------------------------------
template <int C>
__global__ void k_agg_init(const float* __restrict__ h, const float* __restrict__ dinv,
                           const float* __restrict__ bias, float* __restrict__ agg,
                           int n)
{
    long long i = (long long)blockIdx.x * blockDim.x + threadIdx.x;
    if (i >= (long long)n * C) return;
    int node = (int)(i / C);
    int c    = (int)(i % C);
    float di = dinv[node];
    agg[i] = h[i] * di * di + bias[c];
}

// ---------------------------------------------------------------------------
// edge scatter: agg[dst] += h[src] * dinv[src]*dinv[dst]
// thread per (edge, channel): channel segment per edge is contiguous ->
// coalesced gathers and coalesced f32 atomics (L2-resident on 192MB L2).
// ---------------------------------------------------------------------------
template <int C>
__global__ void k_scatter(const float* __restrict__ h, const int* __restrict__ src,
                          const int* __restrict__ dst, const float* __restrict__ dinv,
                          float* __restrict__ agg, int e)
{
    long long i = (long long)blockIdx.x * blockDim.x + threadIdx.x;
    if (i >= (long long)e * C) return;
    int edge = (int)(i / C);
    int c    = (int)(i % C);
    int s = src[edge];
    int d = dst[edge];
    float w = dinv[s] * dinv[d];
    atomicAdd(&agg[(size_t)d * C + c], h[(size_t)s * C + c] * w);
}

// ---------------------------------------------------------------------------
extern "C" void kernel_launch(void* const* d_in, const int* in_sizes, int n_in,
                              void* d_out, int out_size, void* d_ws, size_t ws_size,
                              hipStream_t stream)
{
    (void)n_in; (void)out_size; (void)ws_size;

    const float* x  = (const float*)d_in[0];
    const int*   ei = (const int*)  d_in[1];
    const float* W1 = (const float*)d_in[2];
    const float* b1 = (const float*)d_in[3];
    const float* W2 = (const float*)d_in[4];
    const float* b2 = (const float*)d_in[5];
    float* out = (float*)d_out;

    const int n = in_sizes[0] / C_IN;   // 100000 (multiple of 16)
    const int e = in_sizes[1] / 2;      // 1600000
    const int* src = ei;
    const int* dst = ei + e;

    // workspace layout (floats): dinv[n] | h (n*64, reused for h2) | agg1 (n*64)
    float* dinv = (float*)d_ws;
    float* h    = dinv + n;
    float* agg1 = h + (size_t)n * C_HID;

    const int B = 256;
    const int n_tiles = n / 16;

    // 1) symmetric normalization
    k_set_ones  <<<(n + B - 1) / B, B, 0, stream>>>(dinv, n);
    k_deg_accum <<<(e + B - 1) / B, B, 0, stream>>>(dst, dinv, e);
    k_deg_to_dinv<<<(n + B - 1) / B, B, 0, stream>>>(dinv, n);

    // 2) h1 = x @ W1   (f32 WMMA)
    k_gemm1_wmma<<<n_tiles, 128, 0, stream>>>(x, W1, h);

    // 3) layer-1 aggregate: self-loop + bias init, then edge scatter
    {
        long long tot = (long long)n * C_HID;
        k_agg_init<C_HID><<<(unsigned)((tot + B - 1) / B), B, 0, stream>>>(h, dinv, b1, agg1, n);
        long long st = (long long)e * C_HID;
        k_scatter<C_HID><<<(unsigned)((st + B - 1) / B), B, 0, stream>>>(h, src, dst, dinv, agg1, e);
    }

    // 4) h2 = relu(agg1) @ W2  (f32 WMMA, ReLU fused) -> reuse h buffer
    k_gemm2_wmma<<<(n_tiles + 7) / 8, 256, 0, stream>>>(agg1, W2, h, n_tiles);

    // 5) layer-2 aggregate straight into d_out
    {
        long long tot = (long long)n * C_OUT;
        k_agg_init<C_OUT><<<(unsigned)((tot + B - 1) / B), B, 0, stream>>>(h, dinv, b2, out, n);
        long long st = (long long)e * C_OUT;
        k_scatter<C_OUT><<<(unsigned)((st + B - 1) / B), B, 0, stream>>>(h, src, dst, dinv, out, e);
    }
}